// SoftDTWLoss_52261162057767
// MI455X (gfx1250) — compile-verified
//
#include <hip/hip_runtime.h>

// Soft-DTW, B=64, M=N=512, gamma = 0.01, p = 2.
// One workgroup per batch; 32 waves; wavefront over 16x16 tiles.
// Each tile's cost matrix is built with one v_wmma_f32_16x16x4_f32:
//   (x_j - y_i)^2 = [y_i^2, 1, -2*y_i, 0] . [1, x_j^2, x_j, 0]^T

#define GAMMA_F 0.01f
#define INVG_F  100.0f
#define BIGF    1e30f

typedef __attribute__((ext_vector_type(2))) float v2f;
typedef __attribute__((ext_vector_type(8))) float v8f;

__device__ __forceinline__ float softmin3(float a, float b, float c) {
    float m = fminf(a, fminf(b, c));
    float s = __expf((m - a) * INVG_F) + __expf((m - b) * INVG_F) +
              __expf((m - c) * INVG_F);
    return m - GAMMA_F * __logf(s);
}

constexpr int T  = 512;  // sequence length
constexpr int TT = 32;   // 16x16 tiles per dimension
constexpr int NW = 32;   // waves per workgroup

__global__ __launch_bounds__(1024, 1)
void softdtw_kernel(const float* __restrict__ x, const float* __restrict__ y,
                    float* __restrict__ out) {
    __shared__ float sx[T], sy[T];        // staged series
    __shared__ float Hrow[T];             // bottom-row halo per column
    __shared__ float Hcol[T];             // right-col halo per row
    __shared__ float Hcor[3][TT];         // corner halo, mod-3 rotated by tile-diagonal
    __shared__ float tiles[NW][256];      // one 16x16 cost tile per wave

    const int b    = blockIdx.x;
    const int tid  = threadIdx.x;
    const int lane = tid & 31;
    const int wave = tid >> 5;
    const int hl   = lane & 15;

    if (tid < T) {
        sx[tid]   = x[b * T + tid];
        sy[tid]   = y[b * T + tid];
        Hrow[tid] = BIGF;                 // R[-1][j] = +inf
        Hcol[tid] = BIGF;                 // R[i][-1] = +inf
    }
    __syncthreads();

    for (int td = 0; td < 2 * TT - 1; ++td) {
        int tiStart = td - (TT - 1); if (tiStart < 0) tiStart = 0;
        int tiEnd   = (td < TT) ? td : (TT - 1);
        int count   = tiEnd - tiStart + 1;

        if (wave < count) {               // wave-uniform branch: EXEC stays full
            const int ti = tiStart + wave;
            const int tj = td - ti;
            const int r0 = ti * 16, c0 = tj * 16;

            // ---- 16x16 cost tile via one WMMA (rank-3 factorization) ----
            float yv = sy[r0 + hl];
            float xv = sx[c0 + hl];
            bool lo = (lane < 16);        // lanes 0-15: K=0,1 ; lanes 16-31: K=2,3
            v2f av, bv;
            av.x = lo ? yv * yv   : -2.0f * yv;
            av.y = lo ? 1.0f      : 0.0f;
            bv.x = lo ? 1.0f      : xv;
            bv.y = lo ? xv * xv   : 0.0f;
            v8f cz = {};
            v8f c = __builtin_amdgcn_wmma_f32_16x16x4_f32(
                false, av, false, bv, (short)0, cz, false, false);

            // scatter D tile to LDS: element (m = k + 8*(lane>=16), n = lane&15)
            float* dt = tiles[wave];
            const int mbase = (lane >> 4) * 8;
            #pragma unroll
            for (int k = 0; k < 8; ++k)
                dt[(mbase + k) * 16 + hl] = c[k];

            // ---- halos ----
            float corner;
            if (ti == 0 && tj == 0)      corner = 0.0f;   // R[-1][-1]
            else if (ti == 0 || tj == 0) corner = BIGF;
            else                         corner = Hcor[(td + 1) % 3][tj - 1]; // diag td-2

            float topj   = Hrow[c0 + hl];
            float topjm1 = (hl > 0) ? Hrow[c0 + hl - 1] : BIGF;

            // ---- in-tile skewed DP: lane j owns column j, 31 steps ----
            const int j = lane;           // lanes >= 16 are spectators
            float hist1 = BIGF, hist2 = BIGF;
            for (int t = 0; t < 31; ++t) {
                float nb1 = __shfl_up(hist1, 1, 32);   // neighbor's R[i][j-1]
                float nb2 = __shfl_up(hist2, 1, 32);   // neighbor's R[i-1][j-1]
                int   i   = t - j;
                bool  act = (i >= 0) && (i < 16) && (j < 16);
                int   ic  = i < 0 ? 0 : (i > 15 ? 15 : i);

                float dij  = dt[ic * 16 + hl];
                float lcol = Hcol[r0 + ic];                          // j==0 left
                float ldia = (ic > 0) ? Hcol[r0 + ic - 1] : corner;  // j==0 diag

                float r_up   = (i == 0) ? topj : hist1;
                float r_left = (j == 0) ? lcol : nb1;
                float r_diag = (j == 0) ? ((i == 0) ? corner : ldia)
                                        : ((i == 0) ? topjm1 : nb2);

                float cur = dij + softmin3(r_diag, r_up, r_left);

                if (j == 15 && act) Hcol[r0 + i] = cur;  // right-column halo
                hist2 = act ? hist1 : hist2;
                hist1 = act ? cur   : hist1;
            }
            if (j < 16)  Hrow[c0 + j] = hist1;           // bottom-row halo
            if (j == 15) Hcor[td % 3][tj] = hist1;       // bottom-right corner
        }
        __syncthreads();
    }

    if (tid == 0) out[b] = Hrow[T - 1];   // R[511][511]
}

extern "C" void kernel_launch(void* const* d_in, const int* in_sizes, int n_in,
                              void* d_out, int out_size, void* d_ws, size_t ws_size,
                              hipStream_t stream) {
    const float* x = (const float*)d_in[0];
    const float* y = (const float*)d_in[1];
    float* out = (float*)d_out;
    const int B = in_sizes[0] / T;        // 64
    softdtw_kernel<<<dim3(B), dim3(1024), 0, stream>>>(x, y, out);
}